// GCNAndMLPConcat_32298154065951
// MI455X (gfx1250) — compile-verified
//
#include <hip/hip_runtime.h>

// ---------------------------------------------------------------------------
// GCN + MLP concat classifier for MI455X (gfx1250, wave32, WMMA).
// f32 end-to-end; GEMMs use V_WMMA_F32_16X16X4_F32 (scatter phase dominates
// the roofline, so there is no reason to downcast to bf16/fp8).
// ---------------------------------------------------------------------------

typedef __attribute__((ext_vector_type(2))) float v2f;
typedef __attribute__((ext_vector_type(8))) float v8f;

constexpr int kN  = 100000;   // nodes (== 6250 * 16, tiles exactly)
constexpr int kE  = 1600000;  // edges
constexpr int kXF = 128;      // feature dim
constexpr int kXL = 40;       // label dim
constexpr int kH  = 128;      // hidden
constexpr int kC  = 40;       // classes
constexpr int kKM = kXF + kXL;   // 168, MLP K
constexpr int kKC = 2 * kH;      // 256, classifier K

// ---------------------------- degree pipeline ------------------------------

__global__ void k_init_deg(float* __restrict__ deg) {
  int i = blockIdx.x * blockDim.x + threadIdx.x;
  if (i < kN) deg[i] = 1.0f;   // self loop
}

__global__ void k_deg_accum(const int* __restrict__ eidx, float* __restrict__ deg) {
  int e = blockIdx.x * blockDim.x + threadIdx.x;
  if (e < kE) atomicAdd(deg + eidx[kE + e], 1.0f);   // dst row of edge_index
}

__global__ void k_dinv(float* __restrict__ deg) {
  int i = blockIdx.x * blockDim.x + threadIdx.x;
  if (i < kN) {
    float d = deg[i];
    deg[i] = (d > 0.0f) ? rsqrtf(d) : 0.0f;
  }
}

// ------------------------- GEMM 1: h = xfeat @ W_gcn -----------------------
// One wave per 16x16 output tile. grid = (N/16, H/16), block = 32.

__global__ __launch_bounds__(32)
void k_gemm_gcn(const float* __restrict__ A, const float* __restrict__ B,
                float* __restrict__ Hout) {
  __shared__ float As[16][kXF + 4];   // +4 pad: avoid 16-way bank conflicts
  __shared__ float Bs[kXF][16];

  const int lane = threadIdx.x;
  const int m0 = blockIdx.x * 16;
  const int n0 = blockIdx.y * 16;

  for (int idx = lane; idx < 16 * kXF; idx += 32) {
    int r = idx >> 7, c = idx & 127;
    As[r][c] = A[(size_t)(m0 + r) * kXF + c];
  }
  for (int idx = lane; idx < kXF * 16; idx += 32) {
    int k = idx >> 4, n = idx & 15;
    Bs[k][n] = B[k * kH + n0 + n];
  }
  __syncthreads();

  const int half = lane >> 4;       // 0: K={0,1}, 1: K={2,3}
  const int l    = lane & 15;
  v8f acc = {};

#pragma unroll 4
  for (int kb = 0; kb < kXF; kb += 4) {
    v2f a, b;
    a[0] = As[l][kb + 2 * half + 0];
    a[1] = As[l][kb + 2 * half + 1];
    b[0] = Bs[kb + 2 * half + 0][l];
    b[1] = Bs[kb + 2 * half + 1][l];
    acc = __builtin_amdgcn_wmma_f32_16x16x4_f32(false, a, false, b,
                                                (short)0, acc, false, false);
  }

#pragma unroll
  for (int r = 0; r < 8; ++r)
    Hout[(size_t)(m0 + half * 8 + r) * kH + (n0 + l)] = acc[r];
}

// --------------- self-loop contribution: comb[:, :128] init ----------------

__global__ void k_selfloop(const float* __restrict__ h,
                           const float* __restrict__ dinv,
                           float* __restrict__ comb) {
  int t = blockIdx.x * blockDim.x + threadIdx.x;
  if (t < kN * kH) {
    int i = t >> 7, c = t & 127;
    float di = dinv[i];
    comb[(size_t)i * kKC + c] = h[(size_t)i * kH + c] * di * di;
  }
}

// --------------- edge scatter: comb[dst, :128] += h[src]*norm --------------
// 1 thread per (edge, 4-feature chunk): float4 gather + 4 f32 atomics.

__global__ void k_edge_scatter(const int* __restrict__ eidx,
                               const float* __restrict__ h,
                               const float* __restrict__ dinv,
                               float* __restrict__ comb) {
  long long t = (long long)blockIdx.x * blockDim.x + threadIdx.x;
  if (t >= (long long)kE * 32) return;
  int e = (int)(t >> 5);
  int q = (int)(t & 31);
  int s = eidx[e];
  int d = eidx[kE + e];
  float nm = dinv[s] * dinv[d];
  const float4 hv = *reinterpret_cast<const float4*>(h + (size_t)s * kH + q * 4);
  float* p = comb + (size_t)d * kKC + q * 4;
  atomicAdd(p + 0, hv.x * nm);
  atomicAdd(p + 1, hv.y * nm);
  atomicAdd(p + 2, hv.z * nm);
  atomicAdd(p + 3, hv.w * nm);
}

// ------- GEMM 2: comb[:,128:] = relu(concat(xfeat,xlabel) @ W_mlp + b) -----

__global__ __launch_bounds__(32)
void k_gemm_mlp(const float* __restrict__ xf, const float* __restrict__ xl,
                const float* __restrict__ W, const float* __restrict__ bias,
                float* __restrict__ comb) {
  __shared__ float As[16][kKM + 4];   // 168 + 4
  __shared__ float Bs[kKM][16];

  const int lane = threadIdx.x;
  const int m0 = blockIdx.x * 16;
  const int n0 = blockIdx.y * 16;

  for (int idx = lane; idx < 16 * kKM; idx += 32) {
    int r = idx / kKM, c = idx % kKM;
    As[r][c] = (c < kXF) ? xf[(size_t)(m0 + r) * kXF + c]
                         : xl[(size_t)(m0 + r) * kXL + (c - kXF)];
  }
  for (int idx = lane; idx < kKM * 16; idx += 32) {
    int k = idx >> 4, n = idx & 15;
    Bs[k][n] = W[k * kH + n0 + n];
  }
  __syncthreads();

  const int half = lane >> 4;
  const int l    = lane & 15;
  v8f acc = {};

#pragma unroll 2
  for (int kb = 0; kb < kKM; kb += 4) {
    v2f a, b;
    a[0] = As[l][kb + 2 * half + 0];
    a[1] = As[l][kb + 2 * half + 1];
    b[0] = Bs[kb + 2 * half + 0][l];
    b[1] = Bs[kb + 2 * half + 1][l];
    acc = __builtin_amdgcn_wmma_f32_16x16x4_f32(false, a, false, b,
                                                (short)0, acc, false, false);
  }

  const float bn = bias[n0 + l];
#pragma unroll
  for (int r = 0; r < 8; ++r) {
    float v = acc[r] + bn;
    v = v > 0.0f ? v : 0.0f;
    comb[(size_t)(m0 + half * 8 + r) * kKC + kH + (n0 + l)] = v;
  }
}

// ----------------- GCN epilogue: comb[:, :128] = relu(x + b) ---------------

__global__ void k_gcn_bias_relu(float* __restrict__ comb, const float* __restrict__ bg) {
  int t = blockIdx.x * blockDim.x + threadIdx.x;
  if (t < kN * kH) {
    int i = t >> 7, c = t & 127;
    float v = comb[(size_t)i * kKC + c] + bg[c];
    comb[(size_t)i * kKC + c] = v > 0.0f ? v : 0.0f;
  }
}

// ---------------- GEMM 3: out = comb @ W_cls + b_cls (N=40 -> pad 48) ------

__global__ __launch_bounds__(32)
void k_gemm_cls(const float* __restrict__ comb, const float* __restrict__ W,
                const float* __restrict__ bias, float* __restrict__ out) {
  __shared__ float As[16][kKC + 4];   // 256 + 4
  __shared__ float Bs[kKC][16];

  const int lane = threadIdx.x;
  const int m0 = blockIdx.x * 16;
  const int n0 = blockIdx.y * 16;    // 0, 16, 32

  for (int idx = lane; idx < 16 * kKC; idx += 32) {
    int r = idx >> 8, c = idx & 255;
    As[r][c] = comb[(size_t)(m0 + r) * kKC + c];
  }
  for (int idx = lane; idx < kKC * 16; idx += 32) {
    int k = idx >> 4, n = idx & 15;
    Bs[k][n] = (n0 + n < kC) ? W[k * kC + n0 + n] : 0.0f;   // zero-pad cols
  }
  __syncthreads();

  const int half = lane >> 4;
  const int l    = lane & 15;
  v8f acc = {};

#pragma unroll 4
  for (int kb = 0; kb < kKC; kb += 4) {
    v2f a, b;
    a[0] = As[l][kb + 2 * half + 0];
    a[1] = As[l][kb + 2 * half + 1];
    b[0] = Bs[kb + 2 * half + 0][l];
    b[1] = Bs[kb + 2 * half + 1][l];
    acc = __builtin_amdgcn_wmma_f32_16x16x4_f32(false, a, false, b,
                                                (short)0, acc, false, false);
  }

  const int col = n0 + l;
  if (col < kC) {
    const float bn = bias[col];
#pragma unroll
    for (int r = 0; r < 8; ++r)
      out[(size_t)(m0 + half * 8 + r) * kC + col] = acc[r] + bn;
  }
}

// ---------------------------------------------------------------------------

extern "C" void kernel_launch(void* const* d_in, const int* in_sizes, int n_in,
                              void* d_out, int out_size, void* d_ws, size_t ws_size,
                              hipStream_t stream) {
  (void)in_sizes; (void)n_in; (void)out_size; (void)ws_size;

  const float* xfeat  = (const float*)d_in[0];   // [N, 128]
  const float* xlabel = (const float*)d_in[1];   // [N, 40]
  const int*   eidx   = (const int*)d_in[2];     // [2, E]
  const float* W_gcn  = (const float*)d_in[3];   // [128, 128]
  const float* b_gcn  = (const float*)d_in[4];   // [128]
  const float* W_mlp  = (const float*)d_in[5];   // [168, 128]
  const float* b_mlp  = (const float*)d_in[6];   // [128]
  const float* W_cls  = (const float*)d_in[7];   // [256, 40]
  const float* b_cls  = (const float*)d_in[8];   // [40]
  float* out = (float*)d_out;                    // [N, 40]

  // Workspace layout (256B aligned): deg | h | combined
  char* ws = (char*)d_ws;
  const size_t off_deg  = 0;
  const size_t off_h    = (off_deg + (size_t)kN * 4 + 255) & ~(size_t)255;
  const size_t off_comb = (off_h + (size_t)kN * kH * 4 + 255) & ~(size_t)255;
  float* deg  = (float*)(ws + off_deg);    // N, becomes dinv in place
  float* hbuf = (float*)(ws + off_h);      // N x 128
  float* comb = (float*)(ws + off_comb);   // N x 256

  const dim3 blk256(256);

  // 1) degrees -> dinv
  k_init_deg <<<dim3((kN + 255) / 256), blk256, 0, stream>>>(deg);
  k_deg_accum<<<dim3((kE + 255) / 256), blk256, 0, stream>>>(eidx, deg);
  k_dinv     <<<dim3((kN + 255) / 256), blk256, 0, stream>>>(deg);

  // 2) h = xfeat @ W_gcn  (WMMA f32)
  k_gemm_gcn<<<dim3(kN / 16, kH / 16), dim3(32), 0, stream>>>(xfeat, W_gcn, hbuf);

  // 3) self-loop init of comb[:, :128], then edge scatter-add
  k_selfloop<<<dim3((kN * kH) / 256), blk256, 0, stream>>>(hbuf, deg, comb);
  k_edge_scatter<<<dim3((unsigned)(((long long)kE * 32) / 256)), blk256, 0, stream>>>(
      eidx, hbuf, deg, comb);

  // 4) MLP branch into comb[:, 128:]  (WMMA f32, fused concat + bias + relu)
  k_gemm_mlp<<<dim3(kN / 16, kH / 16), dim3(32), 0, stream>>>(
      xfeat, xlabel, W_mlp, b_mlp, comb);

  // 5) GCN bias + relu in place
  k_gcn_bias_relu<<<dim3((kN * kH) / 256), blk256, 0, stream>>>(comb, b_gcn);

  // 6) classifier: out = comb @ W_cls + b_cls  (WMMA f32, N padded to 48)
  k_gemm_cls<<<dim3(kN / 16, 3), dim3(32), 0, stream>>>(comb, W_cls, b_cls, out);
}